// Laplacian_77738908058218
// MI455X (gfx1250) — compile-verified
//
#include <hip/hip_runtime.h>

// Focus stacking, fused single pass, gfx1250 (MI455X, wave32).
//   x: (4, 8, 3, 1024, 1024) f32  ->  out: (4, 3, 1024, 1024) f32
// Per 32x32 output tile (256 threads = 8 wave32):
//   stage 1: reflect-101 grayscale halo tile (40x40) -> LDS
//   stage 2: vertical 5-tap Gaussian as banded 16x16x4 f32 WMMA matmul -> LDS
//   stage 3: horizontal 5-tap Gaussian (VALU) -> LDS
//   stage 4: 5x5 Laplacian (VALU) + running per-pixel argmax over 8 frames
//   final:   gather RGB of winning frame (L2 hits) -> out

#define TILE 32
#define GDIM 40   // TILE + 2*4 halo (blur +-2 then laplacian +-2)
#define VROWS 36  // valid vertically-blurred rows (TILE + 4)
#define VCOLS 40  // valid vertically-blurred cols
#define VPAD 48   // padded LDS extent: 3x16 WMMA tiles store unconditionally
#define BLW 36

typedef __attribute__((ext_vector_type(2))) float v2f;
typedef __attribute__((ext_vector_type(8))) float v8f;

__device__ __forceinline__ int reflect101(int q, int n) {
  q = (q < 0) ? -q : q;
  q = (q >= n) ? (2 * n - 2 - q) : q;
  return q;
}

// Gaussian 5-tap band weight w = [1,4,6,4,1]/16, 0 outside [0,4]
__device__ __forceinline__ float bandw(int d) {
  float v = 0.0f;
  if (d == 0 || d == 4) v = 1.0f / 16.0f;
  if (d == 1 || d == 3) v = 4.0f / 16.0f;
  if (d == 2)           v = 6.0f / 16.0f;
  return v;
}

__launch_bounds__(256)
__global__ void focus_stack_kernel(const float* __restrict__ xin,
                                   float* __restrict__ out) {
  constexpr int S = 8, C = 3, H = 1024, W = 1024;
  constexpr long HW = (long)H * W;

  __shared__ float Gt[GDIM][GDIM];  // reflect-extended grayscale
  __shared__ float Vt[VPAD][VPAD];  // vertically blurred (padded for WMMA stores)
  __shared__ float Bt[VROWS][BLW];  // fully blurred

  const int tid  = threadIdx.x;
  const int lane = tid & 31;
  const int wave = tid >> 5;
  const int hi   = lane >> 4;   // 0: lanes 0-15, 1: lanes 16-31
  const int lm   = lane & 15;
  const int bz   = blockIdx.z;
  const int oy0  = blockIdx.y * TILE;
  const int ox0  = blockIdx.x * TILE;

  // each thread owns 4 output pixels: rows rbase+8j, column col
  const int col   = tid & 31;
  const int rbase = tid >> 5;   // 0..7
  float bestVal[4];
  int   bestIdx[4];
#pragma unroll
  for (int j = 0; j < 4; ++j) { bestVal[j] = -3.402823466e38f; bestIdx[j] = 0; }

#if defined(__gfx1250__) && __has_builtin(__builtin_amdgcn_wmma_f32_16x16x4_f32)
  // Banded A-matrix chunks for the vertical Gaussian, per ISA 16x4 f32 A layout:
  // lane (hi,lm), vgpr t  holds A[lm][t + 2*hi];  A[m][k] = w[4c + k - m].
  v2f aW[5];
#pragma unroll
  for (int c = 0; c < 5; ++c) {
    aW[c].x = bandw(4 * c + 0 + 2 * hi - lm);
    aW[c].y = bandw(4 * c + 1 + 2 * hi - lm);
  }
#endif

  for (int s = 0; s < S; ++s) {
    __syncthreads();  // prev-iteration LDS readers done before overwriting

    // ---- stage 1: grayscale halo tile with reflect-101 borders ----
    const long frame = ((long)(bz * S + s)) * C * HW;
    for (int idx = tid; idx < GDIM * GDIM; idx += 256) {
      int gy = idx / GDIM, gx = idx - gy * GDIM;
      int yy = reflect101(oy0 - 4 + gy, H);
      int xx = reflect101(ox0 - 4 + gx, W);
      long p = frame + (long)yy * W + xx;
      float r = xin[p], g = xin[p + HW], b = xin[p + 2 * HW];
      Gt[gy][gx] = (r + g + b) * (1.0f / 3.0f);
    }
    __syncthreads();

    // ---- stage 2: vertical 5-tap Gaussian as banded f32 WMMA matmul ----
    // V(36x40) = W_band(36x40) * G(40x40); 3x3 grid of 16x16 tiles, K in 5
    // chunks of 4 covering the 20-wide band. Wave-uniform tile loop => EXEC
    // is all-ones for every WMMA. B reads clamp into Gt (finite values whose
    // weights are zero for all valid rows); stores land in the 48x48 padded
    // Vt so they are unconditional straight-line ds_stores.
#if defined(__gfx1250__) && __has_builtin(__builtin_amdgcn_wmma_f32_16x16x4_f32)
    for (int t = wave; t < 9; t += 8) {
      int my = t / 3, nx = t - my * 3;
      int y0 = my * 16, x0 = nx * 16;
      int bcol = x0 + lm; if (bcol > GDIM - 1) bcol = GDIM - 1;
      v8f acc = {};
#pragma unroll
      for (int c = 0; c < 5; ++c) {
        // B layout: vgpr t holds row K = t + 2*hi of the 4x16 chunk.
        int r0 = y0 + 4 * c + 2 * hi;
        int ra = (r0     < GDIM) ? r0       : (GDIM - 1);
        int rb = (r0 + 1 < GDIM) ? (r0 + 1) : (GDIM - 1);
        v2f bm;
        bm.x = Gt[ra][bcol];
        bm.y = Gt[rb][bcol];
        acc = __builtin_amdgcn_wmma_f32_16x16x4_f32(false, aW[c], false, bm,
                                                    (short)0, acc, false, false);
      }
      // C/D layout: vgpr i, lane -> row y0 + i + 8*hi, col x0 + lm
#pragma unroll
      for (int i = 0; i < 8; ++i)
        Vt[y0 + i + 8 * hi][x0 + lm] = acc[i];
    }
#else
    for (int idx = tid; idx < VROWS * VCOLS; idx += 256) {
      int y = idx / VCOLS, x = idx - (idx / VCOLS) * VCOLS;
      float a  = Gt[y][x] + Gt[y + 4][x];
      float b2 = Gt[y + 1][x] + Gt[y + 3][x];
      Vt[y][x] = a * (1.0f / 16.0f) + b2 * (4.0f / 16.0f) + Gt[y + 2][x] * (6.0f / 16.0f);
    }
#endif
    __syncthreads();

    // ---- stage 3: horizontal 5-tap Gaussian ----
    for (int idx = tid; idx < VROWS * BLW; idx += 256) {
      int y = idx / BLW, x = idx - (idx / BLW) * BLW;
      float a  = Vt[y][x] + Vt[y][x + 4];
      float b2 = Vt[y][x + 1] + Vt[y][x + 3];
      Bt[y][x] = a * (1.0f / 16.0f) + b2 * (4.0f / 16.0f) + Vt[y][x + 2] * (6.0f / 16.0f);
    }
    __syncthreads();

    // ---- stage 4: 5x5 Laplacian (K = S (x) D2 + D2 (x) S) + argmax ----
    constexpr float LK[5][5] = {
        { 2.f,  4.f,   4.f,  4.f,  2.f},
        { 4.f,  0.f,  -8.f,  0.f,  4.f},
        { 4.f, -8.f, -24.f, -8.f,  4.f},
        { 4.f,  0.f,  -8.f,  0.f,  4.f},
        { 2.f,  4.f,   4.f,  4.f,  2.f}};
#pragma unroll
    for (int j = 0; j < 4; ++j) {
      int row = rbase + 8 * j;
      float acc = 0.0f;
#pragma unroll
      for (int i = 0; i < 5; ++i) {
#pragma unroll
        for (int k = 0; k < 5; ++k) {
          if (LK[i][k] != 0.0f) acc += LK[i][k] * Bt[row + i][col + k];
        }
      }
      if (acc > bestVal[j]) { bestVal[j] = acc; bestIdx[j] = s; }  // strict >: first max wins
    }
  }

  // ---- gather RGB of the sharpest frame (tile region is hot in L2) ----
#pragma unroll
  for (int j = 0; j < 4; ++j) {
    int yg = oy0 + rbase + 8 * j;
    int xg = ox0 + col;
    long base_in  = ((long)(bz * S + bestIdx[j])) * C * HW + (long)yg * W + xg;
    long base_out = (long)bz * C * HW + (long)yg * W + xg;
#pragma unroll
    for (int c = 0; c < C; ++c)
      out[base_out + (long)c * HW] = xin[base_in + (long)c * HW];
  }
}

extern "C" void kernel_launch(void* const* d_in, const int* in_sizes, int n_in,
                              void* d_out, int out_size, void* d_ws, size_t ws_size,
                              hipStream_t stream) {
  (void)in_sizes; (void)n_in; (void)d_ws; (void)ws_size; (void)out_size;
  const float* x = (const float*)d_in[0];
  float* out = (float*)d_out;
  dim3 grid(1024 / TILE, 1024 / TILE, 4);  // (32, 32, 4)
  focus_stack_kernel<<<grid, 256, 0, stream>>>(x, out);
}